// DampedIMEX1Layer_35862976921823
// MI455X (gfx1250) — compile-verified
//
#include <hip/hip_runtime.h>
#include <hip/hip_bf16.h>
#include <math.h>

typedef __attribute__((ext_vector_type(2))) float v2f;
typedef __attribute__((ext_vector_type(8))) float v8f;

#define EPSV 1e-6f
constexpr int Bsz = 16, L = 4096, H = 128, P = 256;
constexpr int BL  = Bsz * L;   // 65536 rows
constexpr int N2  = 2 * P;     // 512 (re|im concatenated columns)
constexpr int CH  = 64;        // chunk length for scan
constexpr int NC  = L / CH;    // 64 chunks

// ---------------------------------------------------------------------------
// CDNA5 async global->LDS helpers (ASYNCcnt-tracked, ISA ch.10 / 15.18.3)
// ---------------------------------------------------------------------------
__device__ __forceinline__ void async_ld_b32(unsigned ldsoff, const float* g) {
  asm volatile("global_load_async_to_lds_b32 %0, %1, off"
               :: "v"(ldsoff), "v"(g) : "memory");
}
__device__ __forceinline__ void wait_async_le2() {
  asm volatile("s_wait_asynccnt 0x2" ::: "memory");
}
__device__ __forceinline__ void wait_async_le0() {
  asm volatile("s_wait_asynccnt 0x0" ::: "memory");
}

// ---------------------------------------------------------------------------
// Per-channel parameters: M entries, input factors, and M^CH (repeated squaring)
// prm layout: [0..3]=m00,m01,m10,m11  [4..5]=f1,f2  [6..9]=M^CH entries, each [P]
// ---------------------------------------------------------------------------
__global__ void k_params(const float* __restrict__ A_diag,
                         const float* __restrict__ G_diag,
                         const float* __restrict__ dt,
                         float* __restrict__ prm) {
  int p = threadIdx.x;
  float dts   = 1.f / (1.f + expf(-dt[p]));
  float A     = fmaxf(A_diag[p], 0.f);
  float G     = fmaxf(G_diag[p], 0.f);
  float root  = sqrtf(1.f + dts * G);
  float denom = fmaxf(dts * dts, EPSV);
  float A_low  = (2.f + dts * G - 2.f * root) / denom;
  float A_high = (2.f + dts * G + 2.f * root) / denom;
  A = A_low + fmaxf(A - A_low, 0.f) - fmaxf(A - A_high, 0.f);
  float S  = 1.f / (1.f + dts * G);
  float m00 = S, m01 = -A * dts * S, m10 = dts * S, m11 = 1.f - A * dts * dts * S;
  float f1 = dts * S, f2 = dts * dts * S;
  float a = m00, b = m01, c = m10, d = m11;   // M^(2^6) = M^64
#pragma unroll
  for (int i = 0; i < 6; ++i) {
    float na = a * a + b * c, nb = a * b + b * d;
    float nc = c * a + d * c, nd = c * b + d * d;
    a = na; b = nb; c = nc; d = nd;
  }
  prm[0 * P + p] = m00; prm[1 * P + p] = m01; prm[2 * P + p] = m10; prm[3 * P + p] = m11;
  prm[4 * P + p] = f1;  prm[5 * P + p] = f2;
  prm[6 * P + p] = a;   prm[7 * P + p] = b;   prm[8 * P + p] = c;   prm[9 * P + p] = d;
}

// ---------------------------------------------------------------------------
// Pack operands into WMMA-friendly *paired-k* layouts:
//   BtP: GEMM1 B (K=H=128, N=512):  BtP[((k>>1)*N2 + n)*2 + (k&1)] = Bt[k][n]
//        cols: [0,P)=B_re, [P,2P)=B_im
//   CtP: GEMM2 B (K=N2=512, N=128): CtP[((k>>1)*H + h)*2 + (k&1)] = Ct[k][h]
//        rows: [0,P)=+C_re, [P,2P)=-C_im
// One lane then loads both of its K-values with a single b64.
// ---------------------------------------------------------------------------
__global__ void k_pack(const float* __restrict__ Bm, const float* __restrict__ Cm,
                       float* __restrict__ BtP, float* __restrict__ CtP) {
  int i = blockIdx.x * blockDim.x + threadIdx.x;   // H*P threads
  int h = i / P, p = i % P;
  float bre = Bm[(p * H + h) * 2 + 0];
  float bim = Bm[(p * H + h) * 2 + 1];
  BtP[(((h >> 1) * N2) + p)     * 2 + (h & 1)] = bre;
  BtP[(((h >> 1) * N2) + P + p) * 2 + (h & 1)] = bim;
  float cre =  Cm[(h * P + p) * 2 + 0];
  float cim = -Cm[(h * P + p) * 2 + 1];
  CtP[(((p >> 1) * H) + h) * 2 + (p & 1)]              = cre;   // k = p
  CtP[((((P + p) >> 1) * H) + h) * 2 + (p & 1)]        = cim;   // k = P+p
}

// ---------------------------------------------------------------------------
// fp32 WMMA GEMM: Out(BLxN) = Amat(BLxK) * Bpair(paired-k KxN)  [+ D*u epilogue]
// One wave per 16x64 tile; V_WMMA_F32_16X16X4_F32, K stepped by 4.
// fp32 A layout: lanes 0-15 hold (k,k+1), lanes 16-31 hold (k+2,k+3), M=lane%16.
// ---------------------------------------------------------------------------
template <int K, int N, bool FUSE_DU>
__global__ void k_wmma_gemm(const float* __restrict__ Amat,
                            const float* __restrict__ Bpair,
                            float* __restrict__ Out,
                            const float* __restrict__ Uin,
                            const float* __restrict__ Dvec) {
  const int lane  = threadIdx.x;        // 0..31, blockDim=32 -> EXEC all ones
  const int mtile = blockIdx.x;         // 16-row tile
  const int ngrp  = blockIdx.y;         // 64-col group
  const int mrow  = lane & 15;
  const int kq    = lane >> 4;          // 0/1 -> k offset 2*kq
  const float* arow = Amat + (size_t)(mtile * 16 + mrow) * K + 2 * kq;
  const int    ncol = ngrp * 64 + (lane & 15);

  v8f acc0 = {}, acc1 = {}, acc2 = {}, acc3 = {};
  for (int k = 0; k < K; k += 4) {
    v2f a = *(const v2f*)(arow + k);     // A[m][k+2kq], A[m][k+2kq+1] : b64
    const float* bbase = Bpair + (((size_t)(k >> 1) + kq) * N + ncol) * 2;
    v2f b0 = *(const v2f*)(bbase);       // B[k+2kq][n], B[k+2kq+1][n] : b64
    v2f b1 = *(const v2f*)(bbase + 32);
    v2f b2 = *(const v2f*)(bbase + 64);
    v2f b3 = *(const v2f*)(bbase + 96);
    acc0 = __builtin_amdgcn_wmma_f32_16x16x4_f32(false, a, false, b0, (short)0, acc0, false, false);
    acc1 = __builtin_amdgcn_wmma_f32_16x16x4_f32(false, a, false, b1, (short)0, acc1, false, false);
    acc2 = __builtin_amdgcn_wmma_f32_16x16x4_f32(false, a, false, b2, (short)0, acc2, false, false);
    acc3 = __builtin_amdgcn_wmma_f32_16x16x4_f32(false, a, false, b3, (short)0, acc3, false, false);
  }
  // C/D layout: VGPR i: lanes0-15 -> M=i, lanes16-31 -> M=8+i; N=lane%16
  const int r0 = mtile * 16 + (lane >> 4) * 8;
  const int c0 = ngrp * 64 + (lane & 15);
#pragma unroll
  for (int i = 0; i < 8; ++i) {
    size_t row = (size_t)(r0 + i);
    float e0 = acc0[i], e1 = acc1[i], e2 = acc2[i], e3 = acc3[i];
    if (FUSE_DU) {  // N == H path: y += D[h] * u[row,h]
      e0 += Dvec[c0]      * Uin[row * N + c0];
      e1 += Dvec[c0 + 16] * Uin[row * N + c0 + 16];
      e2 += Dvec[c0 + 32] * Uin[row * N + c0 + 32];
      e3 += Dvec[c0 + 48] * Uin[row * N + c0 + 48];
    }
    Out[row * N + c0]      = e0;
    Out[row * N + c0 + 16] = e1;
    Out[row * N + c0 + 32] = e2;
    Out[row * N + c0 + 48] = e3;
  }
}

// ---------------------------------------------------------------------------
// Scan pass 1: per (b,chunk,p) local scan from zero state; emit chunk aggregate.
// One block = one (b,chunk), p = threadIdx.x. The next timestep's 512-float
// slice of Bu is staged into a double-buffered LDS tile with async loads
// (each thread stages & consumes its own element -> only ASYNCcnt gating).
// agg: 4 planes [Bsz*NC*P]: x1r,x2r,x1i,x2i.
// ---------------------------------------------------------------------------
__global__ void k_scan_local(const float* __restrict__ Bu,
                             const float* __restrict__ prm,
                             float* __restrict__ agg) {
  __shared__ float buf[2 * N2];                 // 4 KB double buffer
  const int p     = threadIdx.x;                // 256 threads
  const int chunk = blockIdx.x & (NC - 1);
  const int b     = blockIdx.x >> 6;            // NC = 64
  float m00 = prm[0 * P + p], m01 = prm[1 * P + p];
  float m10 = prm[2 * P + p], m11 = prm[3 * P + p];
  float f1  = prm[4 * P + p], f2  = prm[5 * P + p];
  float x1r = 0.f, x2r = 0.f, x1i = 0.f, x2i = 0.f;
  const float* base = Bu + ((size_t)(b * L + chunk * CH)) * N2 + p;

  const unsigned lds0 = (unsigned)(size_t)buf;  // LDS offset = addr[31:0]
  const unsigned ore0 = lds0 + p * 4u;
  const unsigned oim0 = lds0 + (P + p) * 4u;

  // prime t = 0 into buffer 0
  async_ld_b32(ore0, base);
  async_ld_b32(oim0, base + P);

  for (int t = 0; t < CH; ++t) {
    const int cur = t & 1;
    if (t + 1 < CH) {                           // stage t+1 into other buffer
      const float* g = base + (size_t)(t + 1) * N2;
      const unsigned sl = (unsigned)((cur ^ 1) * N2 * 4);
      async_ld_b32(ore0 + sl, g);
      async_ld_b32(oim0 + sl, g + P);
      wait_async_le2();                         // t's pair has landed
    } else {
      wait_async_le0();
    }
    float br = buf[cur * N2 + p];
    float bi = buf[cur * N2 + P + p];
    float n1r = m00 * x1r + m01 * x2r + f1 * br;
    float n2r = m10 * x1r + m11 * x2r + f2 * br;
    float n1i = m00 * x1i + m01 * x2i + f1 * bi;
    float n2i = m10 * x1i + m11 * x2i + f2 * bi;
    x1r = n1r; x2r = n2r; x1i = n1i; x2i = n2i;
  }
  size_t o = (size_t)blockIdx.x * P + p, stride = (size_t)Bsz * NC * P;
  agg[o] = x1r; agg[o + stride] = x2r; agg[o + 2 * stride] = x1i; agg[o + 3 * stride] = x2i;
}

// ---------------------------------------------------------------------------
// Scan pass 2: per (b,p), sequential over chunks: s_c = M^CH * s_{c-1} + agg_c.
// Stores the state *entering* each chunk into bnd (same layout as agg).
// ---------------------------------------------------------------------------
__global__ void k_scan_chunks(const float* __restrict__ prm,
                              const float* __restrict__ agg,
                              float* __restrict__ bnd) {
  int tid = blockIdx.x * blockDim.x + threadIdx.x;  // Bsz*P threads
  int p = tid & (P - 1);
  int b = tid >> 8;
  float a = prm[6 * P + p], bb = prm[7 * P + p];
  float c = prm[8 * P + p], d  = prm[9 * P + p];
  size_t stride = (size_t)Bsz * NC * P;
  float x1r = 0.f, x2r = 0.f, x1i = 0.f, x2i = 0.f;
  for (int cI = 0; cI < NC; ++cI) {
    size_t o = ((size_t)b * NC + cI) * P + p;
    bnd[o] = x1r; bnd[o + stride] = x2r; bnd[o + 2 * stride] = x1i; bnd[o + 3 * stride] = x2i;
    float g1r = agg[o], g2r = agg[o + stride];
    float g1i = agg[o + 2 * stride], g2i = agg[o + 3 * stride];
    float n1r = a * x1r + bb * x2r + g1r;
    float n2r = c * x1r + d  * x2r + g2r;
    float n1i = a * x1i + bb * x2i + g1i;
    float n2i = c * x1i + d  * x2i + g2i;
    x1r = n1r; x2r = n2r; x1i = n1i; x2i = n2i;
  }
}

// ---------------------------------------------------------------------------
// Scan pass 3: replay local scan from boundary state, overwrite Bu in place
// with x2 (re at col p, im at col P+p) -> becomes Xs for the output GEMM.
// ---------------------------------------------------------------------------
__global__ void k_scan_final(float* __restrict__ Bu,
                             const float* __restrict__ prm,
                             const float* __restrict__ bnd) {
  int tid   = blockIdx.x * blockDim.x + threadIdx.x;
  int p     = tid & (P - 1);
  int chunk = (tid >> 8) & (NC - 1);
  int b     = tid >> 14;
  float m00 = prm[0 * P + p], m01 = prm[1 * P + p];
  float m10 = prm[2 * P + p], m11 = prm[3 * P + p];
  float f1  = prm[4 * P + p], f2  = prm[5 * P + p];
  size_t o = (size_t)tid, stride = (size_t)Bsz * NC * P;
  float x1r = bnd[o], x2r = bnd[o + stride];
  float x1i = bnd[o + 2 * stride], x2i = bnd[o + 3 * stride];
  float* base = Bu + ((size_t)(b * L + chunk * CH)) * N2 + p;
  for (int t = 0; t < CH; ++t) {
    if (t + 1 < CH) __builtin_prefetch(base + (size_t)(t + 1) * N2, 0, 1);
    float br = base[(size_t)t * N2];
    float bi = base[(size_t)t * N2 + P];
    float n1r = m00 * x1r + m01 * x2r + f1 * br;
    float n2r = m10 * x1r + m11 * x2r + f2 * br;
    float n1i = m00 * x1i + m01 * x2i + f1 * bi;
    float n2i = m10 * x1i + m11 * x2i + f2 * bi;
    x1r = n1r; x2r = n2r; x1i = n1i; x2i = n2i;
    base[(size_t)t * N2]     = x2r;   // in-place: Bu -> Xs
    base[(size_t)t * N2 + P] = x2i;
  }
}

// ---------------------------------------------------------------------------
extern "C" void kernel_launch(void* const* d_in, const int* in_sizes, int n_in,
                              void* d_out, int out_size, void* d_ws, size_t ws_size,
                              hipStream_t stream) {
  (void)in_sizes; (void)n_in; (void)out_size; (void)ws_size;
  const float* u  = (const float*)d_in[0];  // (16,4096,128)
  const float* Ad = (const float*)d_in[1];  // (256,)
  const float* Gd = (const float*)d_in[2];  // (256,)
  const float* dt = (const float*)d_in[3];  // (256,)
  const float* Bm = (const float*)d_in[4];  // (256,128,2)
  const float* Cm = (const float*)d_in[5];  // (128,256,2)
  const float* Dv = (const float*)d_in[6];  // (128,)
  float* out = (float*)d_out;               // (16,4096,128)

  char* ws  = (char*)d_ws;
  float* prm = (float*)ws;                                      // 10*256 f
  float* BtP = (float*)(ws + (size_t)(16 << 10));               // 256 KB
  float* CtP = (float*)(ws + (size_t)(16 << 10) + (256 << 10)); // 256 KB
  float* agg = (float*)(ws + ((size_t)1 << 20));                // 4 planes * 1 MB
  float* bnd = (float*)(ws + ((size_t)5 << 20));                // 4 planes * 1 MB
  float* Bu  = (float*)(ws + ((size_t)9 << 20));                // 128 MB

  k_params<<<1, P, 0, stream>>>(Ad, Gd, dt, prm);
  k_pack<<<(H * P) / 256, 256, 0, stream>>>(Bm, Cm, BtP, CtP);

  // Bu = u @ Bt : (65536 x 128) * (128 x 512)
  k_wmma_gemm<H, N2, false><<<dim3(BL / 16, N2 / 64), 32, 0, stream>>>(
      u, BtP, Bu, nullptr, nullptr);

  // chunked linear scan (time-invariant M -> M^64 closed form)
  k_scan_local <<<Bsz * NC, 256, 0, stream>>>(Bu, prm, agg);
  k_scan_chunks<<<(Bsz * P) / 256, 256, 0, stream>>>(prm, agg, bnd);
  k_scan_final <<<(Bsz * NC * P) / 256, 256, 0, stream>>>(Bu, prm, bnd);

  // out = Xs @ Ct + D*u : (65536 x 512) * (512 x 128)
  k_wmma_gemm<N2, H, true><<<dim3(BL / 16, H / 64), 32, 0, stream>>>(
      Bu, CtP, out, u, Dv);
}